// ResidualQuantizer_10565619548578
// MI455X (gfx1250) — compile-verified
//
#include <hip/hip_runtime.h>
#include <math.h>

#define N_ROWS 32768      // 32*1024 residual vectors
#define E_DIM  512
#define N_E    1024
#define LAYERS 4
#define BETA_F 0.25f
#define MT     64         // rows per block
#define LSTR   516        // padded LDS row stride (floats): 516*4B % 256B = 16B -> conflict-free b64

typedef __attribute__((ext_vector_type(2))) float v2f;
typedef __attribute__((ext_vector_type(4))) float v4f;
typedef __attribute__((ext_vector_type(8))) float v8f;

// ---------------- per-code squared norms ----------------
__global__ void cnorm_kernel(const float* __restrict__ cb, float* __restrict__ cnorm) {
  __shared__ float red[128];
  const int b = blockIdx.x;                       // layer*1024 + code
  const float* row = cb + (size_t)b * E_DIM;
  float s = 0.f;
  for (int i = threadIdx.x; i < E_DIM; i += 128) { float v = row[i]; s += v * v; }
  red[threadIdx.x] = s; __syncthreads();
  for (int st = 64; st > 0; st >>= 1) {
    if (threadIdx.x < st) red[threadIdx.x] += red[threadIdx.x + st];
    __syncthreads();
  }
  if (threadIdx.x == 0) cnorm[b] = red[0];
}

// ---------------- fp32 WMMA distance + argmin ----------------
// block: 256 threads (8 waves). Block covers MT=64 rows x all 1024 codes.
// Each wave handles 128 codes as 4 PAIRS of 16-code tiles; every A (residual)
// LDS load is reused for 2 WMMAs, every B (code) global load for 4 WMMAs.
__global__ __launch_bounds__(256) void argmin_kernel(
    const float* __restrict__ z, const float* __restrict__ zqsum,
    const float* __restrict__ cb_l, const float* __restrict__ cnorm_l,
    int* __restrict__ idxbuf, float* __restrict__ out_idx)
{
  __shared__ float shz[MT * LSTR];   // residual tile, padded
  __shared__ float sD[8 * MT];
  __shared__ int   sI[8 * MT];

  const int tid = threadIdx.x;
  const int rowBase = blockIdx.x * MT;

  // Stage residual = z - zqsum into LDS (vec4, coalesced)
  #pragma unroll
  for (int it = 0; it < (MT * (E_DIM / 4)) / 256; ++it) {
    int e   = tid + it * 256;          // vec4 index within tile
    int row = e >> 7;                  // 128 vec4 per row
    int col = e & 127;
    size_t g = (size_t)(rowBase + row) * (E_DIM / 4) + col;
    v4f zz = ((const v4f*)z)[g];
    v4f qq = ((const v4f*)zqsum)[g];
    v4f r  = zz - qq;
    *(v4f*)&shz[row * LSTR + col * 4] = r;
  }
  __syncthreads();

  const int lane = tid & 31;
  const int wv   = tid >> 5;
  const int lrow = lane & 15;          // A: row-in-tile, B: code-in-tile, C: N = lane&15
  const int koff = (lane >> 4) << 1;   // K offset 0 or 2 per ISA 16x4 fp32 layout

  float bD[4][8];
  int   bI[4][8];
  #pragma unroll
  for (int rt = 0; rt < 4; ++rt)
    #pragma unroll
    for (int v = 0; v < 8; ++v) { bD[rt][v] = 3.0e38f; bI[rt][v] = 0; }

  for (int cp = 0; cp < 4; ++cp) {                 // pair of 16-code tiles
    const int n0 = wv * 128 + cp * 32 + lrow;      // first code of pair (this lane)
    const int n1 = n0 + 16;                        // second code of pair
    const float* cptr0 = cb_l + (size_t)n0 * E_DIM + koff;
    const float* cptr1 = cb_l + (size_t)n1 * E_DIM + koff;
    v8f acc0[4] = {};
    v8f acc1[4] = {};
    for (int k = 0; k < E_DIM; k += 4) {
      v2f b0 = *(const v2f*)(cptr0 + k);
      v2f b1 = *(const v2f*)(cptr1 + k);
      #pragma unroll
      for (int rt = 0; rt < 4; ++rt) {
        v2f a = *(const v2f*)&shz[(rt * 16 + lrow) * LSTR + k + koff];
        acc0[rt] = __builtin_amdgcn_wmma_f32_16x16x4_f32(
            false, a, false, b0, (short)0, acc0[rt], false, false);
        acc1[rt] = __builtin_amdgcn_wmma_f32_16x16x4_f32(
            false, a, false, b1, (short)0, acc1[rt], false, false);
      }
    }
    const float cn0 = cnorm_l[n0];
    const float cn1 = cnorm_l[n1];
    // n0 < n1: update with n0 first to keep first-index argmin semantics
    #pragma unroll
    for (int rt = 0; rt < 4; ++rt)
      #pragma unroll
      for (int v = 0; v < 8; ++v) {
        float d0 = cn0 - 2.0f * acc0[rt][v];
        if (d0 < bD[rt][v]) { bD[rt][v] = d0; bI[rt][v] = n0; }
        float d1 = cn1 - 2.0f * acc1[rt][v];
        if (d1 < bD[rt][v]) { bD[rt][v] = d1; bI[rt][v] = n1; }
      }
  }

  // reduce across the 16 lanes sharing each row (xor butterfly stays within halves)
  #pragma unroll
  for (int m = 1; m <= 8; m <<= 1) {
    #pragma unroll
    for (int rt = 0; rt < 4; ++rt)
      #pragma unroll
      for (int v = 0; v < 8; ++v) {
        float od = __shfl_xor(bD[rt][v], m, 32);
        int   oi = __shfl_xor(bI[rt][v], m, 32);
        if (od < bD[rt][v] || (od == bD[rt][v] && oi < bI[rt][v])) {
          bD[rt][v] = od; bI[rt][v] = oi;
        }
      }
  }

  if (lrow == 0) {                      // lanes 0 and 16 write (half-dependent rows)
    const int half = lane >> 4;
    #pragma unroll
    for (int rt = 0; rt < 4; ++rt)
      #pragma unroll
      for (int v = 0; v < 8; ++v) {
        int r = rt * 16 + half * 8 + v;
        sD[wv * MT + r] = bD[rt][v];
        sI[wv * MT + r] = bI[rt][v];
      }
  }
  __syncthreads();

  if (tid < MT) {                       // final reduce across 8 waves (ascending code ranges)
    float bd = sD[tid]; int bi = sI[tid];
    #pragma unroll
    for (int w = 1; w < 8; ++w) {
      float d = sD[w * MT + tid]; int i = sI[w * MT + tid];
      if (d < bd || (d == bd && i < bi)) { bd = d; bi = i; }
    }
    int grow = rowBase + tid;
    idxbuf[grow]  = bi;
    out_idx[grow] = (float)bi;
  }
}

// ---------------- gather, loss, z_q_sum update, histogram ----------------
__global__ __launch_bounds__(256) void update_kernel(
    const float* __restrict__ z, const float* __restrict__ cb_l,
    const int* __restrict__ idxbuf, float* __restrict__ zqsum,
    float* __restrict__ blockloss, int* __restrict__ counts_l)
{
  const int tid = blockIdx.x * 256 + threadIdx.x;   // 4096*256 threads
  float lsum = 0.f;
  #pragma unroll
  for (int it = 0; it < 4; ++it) {
    int e   = tid + it * (4096 * 256);   // vec4 index, 4,194,304 total
    int row = e >> 7;
    int col = e & 127;
    int idx = idxbuf[row];
    v4f c  = ((const v4f*)cb_l)[(size_t)idx * 128 + col];
    v4f zz = ((const v4f*)z)[e];
    v4f qq = ((const v4f*)zqsum)[e];
    v4f diff = c - (zz - qq);            // z_q - residual
    lsum += diff.x * diff.x + diff.y * diff.y + diff.z * diff.z + diff.w * diff.w;
    ((v4f*)zqsum)[e] = qq + c;
    if (col == 0) atomicAdd(&counts_l[idx], 1);   // integer atomic: deterministic
  }
  __shared__ float red[256];
  red[threadIdx.x] = lsum; __syncthreads();
  for (int st = 128; st > 0; st >>= 1) {
    if (threadIdx.x < st) red[threadIdx.x] += red[threadIdx.x + st];
    __syncthreads();
  }
  if (threadIdx.x == 0) blockloss[blockIdx.x] = red[0];
}

__global__ void loss_finalize_kernel(const float* __restrict__ blockloss,
                                     float* __restrict__ out_loss) {
  __shared__ float red[256];
  float s = 0.f;
  for (int i = threadIdx.x; i < 4096; i += 256) s += blockloss[i];  // fixed order
  red[threadIdx.x] = s; __syncthreads();
  for (int st = 128; st > 0; st >>= 1) {
    if (threadIdx.x < st) red[threadIdx.x] += red[threadIdx.x + st];
    __syncthreads();
  }
  if (threadIdx.x == 0) out_loss[0] += red[0] * (BETA_F / 16777216.0f);
}

__global__ void perp_kernel(const int* __restrict__ counts_l,
                            float* __restrict__ out_perp_l) {
  __shared__ float red[1024];
  float e = (float)counts_l[threadIdx.x] * (1.0f / 32768.0f);
  red[threadIdx.x] = e * logf(e + 1e-10f);
  __syncthreads();
  for (int st = 512; st > 0; st >>= 1) {
    if (threadIdx.x < st) red[threadIdx.x] += red[threadIdx.x + st];
    __syncthreads();
  }
  if (threadIdx.x == 0) out_perp_l[0] = expf(-red[0]);
}

extern "C" void kernel_launch(void* const* d_in, const int* in_sizes, int n_in,
                              void* d_out, int out_size, void* d_ws, size_t ws_size,
                              hipStream_t stream) {
  const float* z  = (const float*)d_in[0];   // [32,1024,512]
  const float* cb = (const float*)d_in[1];   // [4,1024,512]

  float* out      = (float*)d_out;
  float* out_loss = out;                                     // [1]
  float* out_zq   = out + 1;                                 // [16777216]
  float* out_idx  = out + 1 + (size_t)N_ROWS * E_DIM;        // [4*32768] as float
  float* out_perp = out_idx + (size_t)LAYERS * N_ROWS;       // [4]

  int*   idxbuf    = (int*)d_ws;                  // 32768
  int*   counts    = idxbuf + N_ROWS;             // 4*1024
  float* cnorm     = (float*)(counts + LAYERS * N_E);   // 4*1024
  float* blockloss = cnorm + LAYERS * N_E;        // 4096

  hipMemsetAsync(d_out, 0, (1 + (size_t)N_ROWS * E_DIM) * sizeof(float), stream);
  hipMemsetAsync(counts, 0, LAYERS * N_E * sizeof(int), stream);

  cnorm_kernel<<<LAYERS * N_E, 128, 0, stream>>>(cb, cnorm);

  for (int l = 0; l < LAYERS; ++l) {
    const float* cb_l = cb + (size_t)l * N_E * E_DIM;
    argmin_kernel<<<N_ROWS / MT, 256, 0, stream>>>(
        z, out_zq, cb_l, cnorm + l * N_E, idxbuf, out_idx + (size_t)l * N_ROWS);
    update_kernel<<<4096, 256, 0, stream>>>(
        z, cb_l, idxbuf, out_zq, blockloss, counts + l * N_E);
    loss_finalize_kernel<<<1, 256, 0, stream>>>(blockloss, out_loss);
    perp_kernel<<<1, 1024, 0, stream>>>(counts + l * N_E, out_perp + l);
  }
}